// GatingNetwork_72060961292584
// MI455X (gfx1250) — compile-verified
//
#include <hip/hip_runtime.h>
#include <math.h>

typedef __attribute__((ext_vector_type(2))) float v2f;
typedef __attribute__((ext_vector_type(8))) float v8f;

#define B_ 32
#define T_ 2048
#define C_ 2048
#define E_ 64
#define TCH 8                       // T chunks for partial reduction
#define ROWS_PER_CHUNK (T_ / TCH)   // 256
#define CT 2                        // column tiles of 1024 floats (256 thr * float4)

#define EPSV 1e-12f
#define NEG_MAXV 3.4028234663852886e38f

// workspace layout (float offsets)
#define WS_PARTIAL 0
#define WS_SEQREP (WS_PARTIAL + B_ * TCH * C_)   // + 524288
#define WS_ROWN   (WS_SEQREP + B_ * C_)          // + 65536
#define WS_COLN   (WS_ROWN + B_)                 // + 32
#define WS_PROBS  (WS_COLN + E_)                 // + 64
#define WS_MASKSEQ (WS_PROBS + B_ * E_)          // + 2048

// output layout (float offsets): (probs[B,T,E], fallback_count, logits[B,E], mask[B,T,E])
static const size_t OUT_PROBS  = 0;
static const size_t OUT_FBC    = (size_t)B_ * T_ * E_;       // 4194304
static const size_t OUT_LOGITS = OUT_FBC + 1;
static const size_t OUT_MASK   = OUT_LOGITS + (size_t)B_ * E_;

// ---------------- Stage 1: streaming partial reduction over T ----------------
// grid = B * CT * TCH blocks, 256 threads; each thread owns 4 consecutive
// columns (float4, fully coalesced b128 loads) and sums ROWS_PER_CHUNK rows.
__global__ __launch_bounds__(256) void k_reduce(const float* __restrict__ hs,
                                                float* __restrict__ partial) {
    int bid = blockIdx.x;
    int b  = bid / (CT * TCH);
    int r  = bid % (CT * TCH);
    int ct = r / TCH;
    int tc = r % TCH;
    int cbase = ct * 1024 + threadIdx.x * 4;
    const float4* src = (const float4*)(hs + (size_t)b * T_ * C_ +
                                        (size_t)tc * ROWS_PER_CHUNK * C_ + cbase);
    float4 acc; acc.x = 0.f; acc.y = 0.f; acc.z = 0.f; acc.w = 0.f;
#pragma unroll 4
    for (int t = 0; t < ROWS_PER_CHUNK; ++t) {
        float4 x = *src;
        acc.x += x.x; acc.y += x.y; acc.z += x.z; acc.w += x.w;
        src += C_ / 4;
    }
    *(float4*)(partial + (size_t)(b * TCH + tc) * C_ + cbase) = acc;
}

// ---------------- Stage 2: finalize mean + row L2 norms ----------------
// one block per batch row b; deterministic fixed-order reduction
__global__ __launch_bounds__(256) void k_rowfin(const float* __restrict__ partial,
                                                float* __restrict__ seq_rep,
                                                float* __restrict__ rown) {
    int b = blockIdx.x;
    __shared__ float red[256];
    float sq = 0.f;
    for (int i = 0; i < C_ / 256; ++i) {
        int c = i * 256 + threadIdx.x;
        float s = 0.f;
        for (int tc = 0; tc < TCH; ++tc)
            s += partial[(size_t)(b * TCH + tc) * C_ + c];
        float m = s * (1.0f / (float)T_);
        seq_rep[(size_t)b * C_ + c] = m;
        sq += m * m;
    }
    red[threadIdx.x] = sq;
    __syncthreads();
    for (int off = 128; off > 0; off >>= 1) {
        if (threadIdx.x < off) red[threadIdx.x] += red[threadIdx.x + off];
        __syncthreads();
    }
    if (threadIdx.x == 0) rown[b] = sqrtf(red[0]);
}

// ---------------- Stage 3: sim_matrix column L2 norms ----------------
__global__ __launch_bounds__(256) void k_colfin(const float* __restrict__ sim,
                                                float* __restrict__ coln) {
    int e = blockIdx.x;
    __shared__ float red[256];
    float sq = 0.f;
    for (int c = threadIdx.x; c < C_; c += 256) {
        float v = sim[(size_t)c * E_ + e];
        sq += v * v;
    }
    red[threadIdx.x] = sq;
    __syncthreads();
    for (int off = 128; off > 0; off >>= 1) {
        if (threadIdx.x < off) red[threadIdx.x] += red[threadIdx.x + off];
        __syncthreads();
    }
    if (threadIdx.x == 0) coln[e] = sqrtf(red[0]);
}

// ---------------- Stage 4: WMMA GEMM (fp32 16x16x4) + full gating ----------------
// Single 256-thread block (8 wave32). Wave w owns tile (Mtile=w>>2, Ntile=w&3).
// G[32][64] = seq_rep[32,2048] x sim[2048,64]; divide by norms after.
__global__ __launch_bounds__(256) void k_gate(const float* __restrict__ seq_rep,
                                              const float* __restrict__ sim,
                                              const float* __restrict__ gates,
                                              const float* __restrict__ rown,
                                              const float* __restrict__ coln,
                                              float* __restrict__ probs_seq,
                                              float* __restrict__ mask_seq,
                                              float* __restrict__ out_logits,
                                              float* __restrict__ out_fbc) {
    __shared__ float Gs[B_ * E_];
    __shared__ float inact[B_];
    int tid  = threadIdx.x;
    int wave = tid >> 5;
    int lane = tid & 31;
    int mt = wave >> 2, nt = wave & 3;
    int lh = lane & 15;
    int hi2 = (lane >> 4) * 2;           // lanes 0-15 -> K {0,1}; lanes 16-31 -> K {2,3}
    int m = mt * 16 + lh;
    int n = nt * 16 + lh;
    const float* arow = seq_rep + (size_t)m * C_;
    v8f acc = {};
    for (int k0 = 0; k0 < C_; k0 += 4) {
        int ka = k0 + hi2;
        v2f a;  a.x  = arow[ka];                       a.y  = arow[ka + 1];
        v2f bf; bf.x = sim[(size_t)ka * E_ + n];       bf.y = sim[(size_t)(ka + 1) * E_ + n];
        acc = __builtin_amdgcn_wmma_f32_16x16x4_f32(
            /*neg_a=*/false, a, /*neg_b=*/false, bf,
            /*c_mod=*/(short)0, acc, /*reuse_a=*/false, /*reuse_b=*/false);
    }
    // D layout: VGPR j -> M = j (lanes 0-15) or j+8 (lanes 16-31), N = lane%16
    int rbase = mt * 16 + (lane >> 4) * 8;
#pragma unroll
    for (int j = 0; j < 8; ++j) Gs[(rbase + j) * E_ + n] = acc[j];
    __syncthreads();

    if (tid < B_) {
        int b = tid;
        float nr = fmaxf(rown[b], EPSV);
        float lg[E_], gl[E_];
        int mk[E_];
        int active = 0;
        for (int e = 0; e < E_; ++e) {
            float nc  = fmaxf(coln[e], EPSV);
            float aff = Gs[b * E_ + e] / (nr * nc);
            float sg  = 1.0f / (1.0f + expf(-gates[e]));
            float l   = aff - sg;
            lg[e] = l;
            float g = fmaxf(l, 0.0f);
            gl[e] = g;
            mk[e] = (g > 0.0f) ? 1 : 0;
            active += mk[e];
        }
        int isInact = (active == 0) ? 1 : 0;
        if (isInact) {
            // top-k (k = E/2) of lg, tie-break lower index (lax.top_k semantics)
            for (int e = 0; e < E_; ++e) {
                int rank = 0;
                for (int j = 0; j < E_; ++j)
                    rank += (lg[j] > lg[e]) || (lg[j] == lg[e] && j < e);
                if (rank < E_ / 2) mk[e] = 1;
            }
        }
        // masked softmax
        float mx = -NEG_MAXV;
        for (int e = 0; e < E_; ++e) {
            float x = mk[e] ? gl[e] : -NEG_MAXV;
            if (x > mx) mx = x;
        }
        float pr[E_];
        float ssum = 0.f;
        for (int e = 0; e < E_; ++e) {
            float x = mk[e] ? gl[e] : -NEG_MAXV;
            float p = expf(x - mx);
            pr[e] = p; ssum += p;
        }
        float inv = 1.0f / ssum;
        for (int e = 0; e < E_; ++e) {
            probs_seq[b * E_ + e]  = pr[e] * inv;
            mask_seq[b * E_ + e]   = (float)mk[e];
            out_logits[b * E_ + e] = lg[e];
        }
        inact[b] = (float)isInact;
    }
    __syncthreads();
    if (tid == 0) {
        float fc = 0.f;
        for (int b = 0; b < B_; ++b) fc += inact[b];
        *out_fbc = fc;
    }
}

// ---------------- Stage 5: broadcast over T with float4 stores ----------------
// grid = B * (T/16); each block copies per-batch 64-float rows to 16 t slots.
__global__ __launch_bounds__(256) void k_bcast(const float* __restrict__ probs_seq,
                                               const float* __restrict__ mask_seq,
                                               float* __restrict__ out_probs,
                                               float* __restrict__ out_mask) {
    __shared__ __align__(16) float sp[E_];
    __shared__ __align__(16) float sm[E_];
    int b   = blockIdx.x >> 7;     // / (T/16 = 128)
    int tch = blockIdx.x & 127;
    if (threadIdx.x < E_)           sp[threadIdx.x]      = probs_seq[b * E_ + threadIdx.x];
    else if (threadIdx.x < 2 * E_)  sm[threadIdx.x - E_] = mask_seq[b * E_ + threadIdx.x - E_];
    __syncthreads();
    int tl = threadIdx.x >> 4;            // 0..15 -> t within chunk
    int v  = (threadIdx.x & 15) * 4;      // 0..60 -> float4 column
    int t  = tch * 16 + tl;
    size_t base = ((size_t)b * T_ + t) * E_ + v;
    float4 p = *(const float4*)(sp + v);
    float4 q = *(const float4*)(sm + v);
    *(float4*)(out_probs + base) = p;
    *(float4*)(out_mask + base)  = q;
}

extern "C" void kernel_launch(void* const* d_in, const int* in_sizes, int n_in,
                              void* d_out, int out_size, void* d_ws, size_t ws_size,
                              hipStream_t stream) {
    const float* hs    = (const float*)d_in[0];
    const float* sim   = (const float*)d_in[1];
    const float* gates = (const float*)d_in[2];
    float* out = (float*)d_out;
    float* ws  = (float*)d_ws;

    float* partial   = ws + WS_PARTIAL;
    float* seq_rep   = ws + WS_SEQREP;
    float* rown      = ws + WS_ROWN;
    float* coln      = ws + WS_COLN;
    float* probs_seq = ws + WS_PROBS;
    float* mask_seq  = ws + WS_MASKSEQ;

    k_reduce<<<B_ * CT * TCH, 256, 0, stream>>>(hs, partial);
    k_rowfin<<<B_, 256, 0, stream>>>(partial, seq_rep, rown);
    k_colfin<<<E_, 256, 0, stream>>>(sim, coln);
    k_gate<<<1, 256, 0, stream>>>(seq_rep, sim, gates, rown, coln,
                                  probs_seq, mask_seq,
                                  out + OUT_LOGITS, out + OUT_FBC);
    k_bcast<<<B_ * (T_ / 16), 256, 0, stream>>>(probs_seq, mask_seq,
                                                out + OUT_PROBS, out + OUT_MASK);
}